// HunterModel_12927851561509
// MI455X (gfx1250) — compile-verified
//
#include <hip/hip_runtime.h>
#include <hip/hip_bf16.h>

typedef float v2f __attribute__((ext_vector_type(2)));
typedef float v8f __attribute__((ext_vector_type(8)));

#define HID 64

// ---------------------------------------------------------------------------
// Degree kernels: deg[i] = 1 (self loop) + #incoming edges; then deg -> rsqrt
// ---------------------------------------------------------------------------
__global__ void gcn_deg_init(float* __restrict__ deg, int n) {
    int i = blockIdx.x * blockDim.x + threadIdx.x;
    if (i < n) deg[i] = 1.0f;  // self loop
}

__global__ void gcn_deg_accum(const long long* __restrict__ dst,
                              float* __restrict__ deg, long long e) {
    long long i = (long long)blockIdx.x * blockDim.x + threadIdx.x;
    if (i < e) atomicAdd(&deg[(int)dst[i]], 1.0f);
}

__global__ void gcn_deg_finalize(float* __restrict__ deg, int n) {
    int i = blockIdx.x * blockDim.x + threadIdx.x;
    if (i < n) {
        float d = deg[i];
        deg[i] = (d > 0.0f) ? rsqrtf(d) : 0.0f;  // becomes dinv
    }
}

// ---------------------------------------------------------------------------
// Fused GEMM + aggregation-init via V_WMMA_F32_16X16X4_F32 (fp32 WMMA):
//   H[i][c]   = (relu?)X[i][:] . W[:][c]            (raw, for edge gather)
//   AGG[i][c] = dinv[i]^2 * H[i][c] + bias[c]       (self-loop + bias seed)
// 128 threads = 4 waves; each wave owns a 16-row x 64-col output stripe.
// A layout (ISA 16x4 f32): lane L: M = L%16; lanes 0-15 carry K=k0+0/1 in
// the two fragment registers, lanes 16-31 carry K=k0+2/3. B mirrors that
// with N = L%16 striped across lanes. C/D: VGPR r -> M = r + (L>=16 ? 8 : 0).
// AGG may alias X: every output row depends only on the same input row, and
// a wave's loads are all consumed (VGPR-resident) before its stores issue.
// APPLY_RELU is a compile-time flag so the non-ReLU path carries zero extra
// VALU work and the ReLU path needs no cndmask select.
// ---------------------------------------------------------------------------
template <bool APPLY_RELU>
__global__ void __launch_bounds__(128)
gcn_gemm_wmma(const float* __restrict__ X, const float* __restrict__ W,
              const float* __restrict__ bias, const float* __restrict__ dinv,
              float* __restrict__ H, float* __restrict__ AGG,
              int N, int K) {
    __shared__ float Ws[HID * HID];  // up to 64x64 = 16 KB
    for (int i = threadIdx.x; i < K * HID; i += blockDim.x) Ws[i] = W[i];
    __syncthreads();

    const int wave = threadIdx.x >> 5;
    const int lane = threadIdx.x & 31;
    const int row0 = (blockIdx.x * 4 + wave) * 16;
    if (row0 >= N) return;  // wave-uniform: EXEC stays all-ones below

    const int m     = lane & 15;        // row within tile (A), col within tile (B)
    const int khalf = (lane >> 4) * 2;  // 0 or 2

    v8f acc0 = {}, acc1 = {}, acc2 = {}, acc3 = {};
    const float* xrow = X + (size_t)(row0 + m) * K;

    for (int k0 = 0; k0 < K; k0 += 4) {
        const int ka = k0 + khalf;
        v2f a;
        a.x = xrow[ka];
        a.y = xrow[ka + 1];
        if (APPLY_RELU) {
            a.x = fmaxf(a.x, 0.0f);
            a.y = fmaxf(a.y, 0.0f);
        }

        const float* w0 = &Ws[ka * HID + m];
        const float* w1 = &Ws[(ka + 1) * HID + m];
        v2f b0, b1, b2, b3;
        b0.x = w0[0];   b0.y = w1[0];
        b1.x = w0[16];  b1.y = w1[16];
        b2.x = w0[32];  b2.y = w1[32];
        b3.x = w0[48];  b3.y = w1[48];

        acc0 = __builtin_amdgcn_wmma_f32_16x16x4_f32(false, a, false, b0, (short)0, acc0, false, false);
        acc1 = __builtin_amdgcn_wmma_f32_16x16x4_f32(false, a, false, b1, (short)0, acc1, false, false);
        acc2 = __builtin_amdgcn_wmma_f32_16x16x4_f32(false, a, false, b2, (short)0, acc2, false, false);
        acc3 = __builtin_amdgcn_wmma_f32_16x16x4_f32(false, a, false, b3, (short)0, acc3, false, false);
    }

    const int rbase = row0 + ((lane >> 4) << 3);
    const int n = lane & 15;
    const float bb0 = bias[n];
    const float bb1 = bias[n + 16];
    const float bb2 = bias[n + 32];
    const float bb3 = bias[n + 48];
#pragma unroll
    for (int r = 0; r < 8; ++r) {
        const int row = rbase + r;
        const float di = dinv[row];
        const float w  = di * di;
        float* hp = H   + (size_t)row * HID + n;
        float* ap = AGG + (size_t)row * HID + n;
        hp[0]  = acc0[r];
        hp[16] = acc1[r];
        hp[32] = acc2[r];
        hp[48] = acc3[r];
        ap[0]  = acc0[r] * w + bb0;
        ap[16] = acc1[r] * w + bb1;
        ap[32] = acc2[r] * w + bb2;
        ap[48] = acc3[r] * w + bb3;
    }
}

// ---------------------------------------------------------------------------
// Edge scatter: 16 threads per edge, float4 gather + 4 float atomics (the
// ~77 MB working set is resident in the 192 MB L2 -> atomics resolve in L2).
// ---------------------------------------------------------------------------
__global__ void gcn_scatter(const long long* __restrict__ src,
                            const long long* __restrict__ dst,
                            const float* __restrict__ dinv,
                            const float* __restrict__ H,
                            float* __restrict__ AGG, long long e) {
    long long idx = (long long)blockIdx.x * blockDim.x + threadIdx.x;
    if (idx >= e * 16) return;
    long long ed = idx >> 4;
    int c = (int)(idx & 15) * 4;
    int s = (int)src[ed];
    int d = (int)dst[ed];
    float coef = dinv[s] * dinv[d];
    float4 v = *(const float4*)(H + (size_t)s * HID + c);
    float* out = AGG + (size_t)d * HID + c;
    atomicAdd(out + 0, v.x * coef);
    atomicAdd(out + 1, v.y * coef);
    atomicAdd(out + 2, v.z * coef);
    atomicAdd(out + 3, v.w * coef);
}

// ---------------------------------------------------------------------------
// Head: node = relu(relu(agg3)[node_index] @ Wp + bp); out = [node@Wa+ba,
// node@Wm+bm, node@Wg+bg, node@Wt+bt]  (4 + 2 + 3 + 10 = 19 floats).
// The layer-3 ReLU is applied here at load time.
// ---------------------------------------------------------------------------
__global__ void __launch_bounds__(64)
gcn_head(const float* __restrict__ X, const int* __restrict__ nodeIdx,
         const float* __restrict__ Wp, const float* __restrict__ bp,
         const float* __restrict__ Wa, const float* __restrict__ ba,
         const float* __restrict__ Wm, const float* __restrict__ bm,
         const float* __restrict__ Wg, const float* __restrict__ bg,
         const float* __restrict__ Wt, const float* __restrict__ bt,
         float* __restrict__ out) {
    __shared__ float h[HID];
    __shared__ float p[HID];
    int t = threadIdx.x;
    int ni = nodeIdx[0];
    h[t] = fmaxf(X[(size_t)ni * HID + t], 0.0f);  // layer-3 ReLU
    __syncthreads();
    float acc = bp[t];
#pragma unroll 8
    for (int k = 0; k < HID; ++k) acc += h[k] * Wp[k * HID + t];
    p[t] = fmaxf(acc, 0.0f);
    __syncthreads();
    if (t < 19) {
        const float* Wh;
        const float* bh;
        int col, ncol;
        if (t < 4)       { Wh = Wa; bh = ba; col = t;     ncol = 4;  }
        else if (t < 6)  { Wh = Wm; bh = bm; col = t - 4; ncol = 2;  }
        else if (t < 9)  { Wh = Wg; bh = bg; col = t - 6; ncol = 3;  }
        else             { Wh = Wt; bh = bt; col = t - 9; ncol = 10; }
        float o = bh[col];
#pragma unroll 8
        for (int k = 0; k < HID; ++k) o += p[k] * Wh[k * ncol + col];
        out[t] = o;
    }
}

// ---------------------------------------------------------------------------
extern "C" void kernel_launch(void* const* d_in, const int* in_sizes, int n_in,
                              void* d_out, int out_size, void* d_ws, size_t ws_size,
                              hipStream_t stream) {
    const float*     x   = (const float*)d_in[0];
    const long long* ei  = (const long long*)d_in[1];
    const int*       nix = (const int*)d_in[2];
    const float* W1 = (const float*)d_in[3];  const float* b1 = (const float*)d_in[4];
    const float* W2 = (const float*)d_in[5];  const float* b2 = (const float*)d_in[6];
    const float* W3 = (const float*)d_in[7];  const float* b3 = (const float*)d_in[8];
    const float* Wp = (const float*)d_in[9];  const float* bp = (const float*)d_in[10];
    const float* Wa = (const float*)d_in[11]; const float* ba = (const float*)d_in[12];
    const float* Wm = (const float*)d_in[13]; const float* bm = (const float*)d_in[14];
    const float* Wg = (const float*)d_in[15]; const float* bg = (const float*)d_in[16];
    const float* Wt = (const float*)d_in[17]; const float* bt = (const float*)d_in[18];

    const int       N = in_sizes[0] / 4;               // 100000
    const long long E = (long long)in_sizes[1] / 2;    // 1600000
    const long long* src = ei;
    const long long* dst = ei + E;

    // workspace layout: dinv[N] | H[N*64] | X/AGG[N*64]
    size_t dinvBytes = (((size_t)N * 4 + 255) / 256) * 256;
    size_t featBytes = (size_t)N * HID * sizeof(float);
    if (ws_size < dinvBytes + 2 * featBytes) return;  // insufficient scratch
    char*  ws   = (char*)d_ws;
    float* dinv = (float*)ws;
    float* bufH = (float*)(ws + dinvBytes);
    float* bufX = (float*)(ws + dinvBytes + featBytes);

    const int TB = 256;
    dim3 blk(TB);
    int  nBlk = (N + TB - 1) / TB;
    long long eBlk = (E + TB - 1) / TB;
    long long sBlk = (E * 16 + TB - 1) / TB;
    int  gBlk = (N / 16 + 3) / 4;  // GEMM: 4 waves/block, 16 rows/wave

    // degrees -> dinv (shared by all three layers)
    gcn_deg_init<<<nBlk, blk, 0, stream>>>(dinv, N);
    gcn_deg_accum<<<(unsigned)eBlk, blk, 0, stream>>>(dst, dinv, E);
    gcn_deg_finalize<<<nBlk, blk, 0, stream>>>(dinv, N);

    // layer 1 (K = 4, reads original x, no input ReLU)
    gcn_gemm_wmma<false><<<gBlk, dim3(128), 0, stream>>>(x, W1, b1, dinv, bufH, bufX, N, 4);
    gcn_scatter<<<(unsigned)sBlk, blk, 0, stream>>>(src, dst, dinv, bufH, bufX, E);

    // layer 2 (K = 64, layer-1 ReLU fused into GEMM load)
    gcn_gemm_wmma<true><<<gBlk, dim3(128), 0, stream>>>(bufX, W2, b2, dinv, bufH, bufX, N, HID);
    gcn_scatter<<<(unsigned)sBlk, blk, 0, stream>>>(src, dst, dinv, bufH, bufX, E);

    // layer 3 (K = 64, layer-2 ReLU fused into GEMM load)
    gcn_gemm_wmma<true><<<gBlk, dim3(128), 0, stream>>>(bufX, W3, b3, dinv, bufH, bufX, N, HID);
    gcn_scatter<<<(unsigned)sBlk, blk, 0, stream>>>(src, dst, dinv, bufH, bufX, E);

    // head (layer-3 ReLU fused into its load)
    gcn_head<<<1, dim3(64), 0, stream>>>(bufX, nix, Wp, bp, Wa, ba, Wm, bm,
                                         Wg, bg, Wt, bt, (float*)d_out);
}